// Net_18906446037087
// MI455X (gfx1250) — compile-verified
//
#include <hip/hip_runtime.h>
#include <hip/hip_bf16.h>

typedef __attribute__((ext_vector_type(2))) float v2f;
typedef __attribute__((ext_vector_type(8))) float v8f;

#define BETA 0.95f
#define THRESH 1.0f
#define NUM_STEPS 255

// ---------------------------------------------------------------------------
// Kernel 1: cur[b] = dot(x[b, :], W[:])  via V_WMMA_F32_16X16X4_F32.
// One wave32 computes a 16-row tile; K consumed 4 at a time (784 = 196 * 4).
// A layout (32-bit A 16x4): lanes 0-15 -> M=lane, VGPR0 = K0/K2 split by lane
// half, VGPR1 = K1/K3. B layout (4x16): N striped across lanes, same K split;
// we broadcast W so every N column of D holds the same row-dot (redundant
// compute, but the kernel is purely HBM-bandwidth bound: 205 MB of x read).
// ---------------------------------------------------------------------------
__global__ void __launch_bounds__(128) gemv_wmma_f32(
    const float* __restrict__ x, const float* __restrict__ W,
    float* __restrict__ cur, int B, int K) {
  const int lane = threadIdx.x & 31;
  const int wave = (blockIdx.x * (blockDim.x >> 5)) + (threadIdx.x >> 5);
  const int row0 = wave * 16;
  if (row0 + 16 > B) return;  // wave-uniform guard: EXEC stays all-1s for WMMA

  const int m     = lane & 15;        // row within tile
  const int khalf = (lane >> 4) * 2;  // K sub-offset: 0 (lanes 0-15) / 2 (16-31)

  const float* __restrict__ xrow = x + (size_t)(row0 + m) * (size_t)K + khalf;
  const float* __restrict__ wrow = W + khalf;

  v8f acc = {};
#pragma unroll 4
  for (int k0 = 0; k0 < K; k0 += 4) {
    v2f a = *(const v2f*)(xrow + k0);   // A: {K=khalf, K=khalf+1} for row m
    v2f b = *(const v2f*)(wrow + k0);   // B: same W pair broadcast to all N
    // D = A*B + C ; emits v_wmma_f32_16x16x4_f32
    acc = __builtin_amdgcn_wmma_f32_16x16x4_f32(
        /*neg_a=*/false, a, /*neg_b=*/false, b,
        /*c_mod=*/(short)0, acc, /*reuse_a=*/false, /*reuse_b=*/false);
  }

  // D layout: lane l, VGPR r -> M = (l<16 ? r : r+8), N = l&15.
  // Column N==0 (lanes 0 and 16) carries all 16 row results.
  if ((lane & 15) == 0) {
    const int mbase = row0 + (lane >> 4) * 8;
#pragma unroll
    for (int r = 0; r < 8; ++r) cur[mbase + r] = acc[r];
  }
}

// ---------------------------------------------------------------------------
// Kernel 2: 255-step LIF recurrence, one thread per batch element.
// mem lives in a register; spk/mem records stream out with non-temporal
// stores (write-once, 134 MB total) coalesced across lanes at each step.
// Reference semantics (reset_mechanism='zero', reset from PREVIOUS mem):
//   mem' = (mem > 1) ? 0 : 0.95*mem + cur ;  spk = (mem' > 1)
// ---------------------------------------------------------------------------
__global__ void __launch_bounds__(256) lif_scan(
    const float* __restrict__ cur, float* __restrict__ spk_rec,
    float* __restrict__ mem_rec, int B) {
  const int b = blockIdx.x * blockDim.x + threadIdx.x;
  if (b >= B) return;

  const float c = cur[b];
  float mem = 0.0f;
#pragma unroll 5
  for (int t = 0; t < NUM_STEPS; ++t) {
    mem = (mem > THRESH) ? 0.0f : fmaf(BETA, mem, c);
    const float spk = (mem > THRESH) ? 1.0f : 0.0f;
    const size_t off = (size_t)t * (size_t)B + (size_t)b;
    __builtin_nontemporal_store(spk, spk_rec + off);
    __builtin_nontemporal_store(mem, mem_rec + off);
  }
}

extern "C" void kernel_launch(void* const* d_in, const int* in_sizes, int n_in,
                              void* d_out, int out_size, void* d_ws, size_t ws_size,
                              hipStream_t stream) {
  const float* x = (const float*)d_in[0];  // [B, K] f32
  const float* W = (const float*)d_in[1];  // [1, K] f32
  const int K = in_sizes[1];               // 784
  const int B = in_sizes[0] / K;           // 65536

  float* cur     = (float*)d_ws;                       // [B] f32 scratch
  float* spk_rec = (float*)d_out;                      // [255, B]
  float* mem_rec = (float*)d_out + (size_t)NUM_STEPS * (size_t)B;

  // GEMV: 16 rows per wave, 4 waves (128 threads) per block -> 64 rows/block.
  const int rows_per_block = 64;
  const int gemv_blocks = (B + rows_per_block - 1) / rows_per_block;
  gemv_wmma_f32<<<gemv_blocks, 128, 0, stream>>>(x, W, cur, B, K);

  // Recurrence: one thread per batch element.
  const int scan_blocks = (B + 255) / 256;
  lif_scan<<<scan_blocks, 256, 0, stream>>>(cur, spk_rec, mem_rec, B);
}